// G_SpaceTime_58299886076385
// MI455X (gfx1250) — compile-verified
//
#include <hip/hip_runtime.h>
#include <hip/hip_bf16.h>

typedef __attribute__((ext_vector_type(2))) float v2f;
typedef __attribute__((ext_vector_type(8))) float v8f;

#define WFENCE() asm volatile("" ::: "memory")

#define XDIM 512
#define CDIM 32
#define NPIX (XDIM * XDIM)

// ---------------------------------------------------------------------------
// Kernel 1: batch-independent bilinear resample of `data` -> feature image
// feat[i][j][c], i = p/512, j = p%512 (matches feats.reshape(X,Y,C))
// ---------------------------------------------------------------------------
__global__ __launch_bounds__(256) void feat_precompute(
    const float* __restrict__ data, const float* __restrict__ lerp_w,
    const int* __restrict__ x0, const int* __restrict__ y0,
    const int* __restrict__ x1, const int* __restrict__ y1,
    float* __restrict__ feat)
{
  int p = blockIdx.x * blockDim.x + threadIdx.x;
  if (p >= NPIX) return;
  float wA = lerp_w[2 * p + 0];
  float wB = lerp_w[2 * p + 1];
  int xa = x0[p], ya = y0[p], xb = x1[p], yb = y1[p];
  float w00 = (1.f - wA) * (1.f - wB);
  float w01 = wA * (1.f - wB);
  float w10 = (1.f - wA) * wB;
  float w11 = wA * wB;
  const float4* p00 = (const float4*)(data + ((size_t)ya * XDIM + xa) * CDIM);
  const float4* p01 = (const float4*)(data + ((size_t)ya * XDIM + xb) * CDIM);
  const float4* p10 = (const float4*)(data + ((size_t)yb * XDIM + xa) * CDIM);
  const float4* p11 = (const float4*)(data + ((size_t)yb * XDIM + xb) * CDIM);
  float4* dst = (float4*)(feat + (size_t)p * CDIM);
#pragma unroll
  for (int q = 0; q < 8; ++q) {
    float4 a = p00[q], b = p01[q], c = p10[q], d = p11[q];
    float4 o;
    o.x = a.x * w00 + b.x * w01 + c.x * w10 + d.x * w11;
    o.y = a.y * w00 + b.y * w01 + c.y * w10 + d.y * w11;
    o.z = a.z * w00 + b.z * w01 + c.z * w10 + d.z * w11;
    o.w = a.w * w00 + b.w * w01 + c.w * w10 + d.w * w11;
    dst[q] = o;
  }
}

// ---------------------------------------------------------------------------
// One 32x32 fully-connected layer on a 16-pixel tile via V_WMMA_F32_16X16X4_F32
// y: per-wave LDS scratch, y[0..511] = 16x32 activations (row-major M*32+K),
//    y[512..527] = LN mu per row, y[528..543] = LN rsigma per row.
// wT: transposed weights in LDS, wT[n*32+k] = W[k][n]
// MODE: 0 = identity A, 1 = sin(layernorm(.)) with (g,bb), 2 = relu
// ---------------------------------------------------------------------------
template <int MODE>
__device__ __forceinline__ void layer32(float* y, const float* wT,
                                        const float* bias, const float* g,
                                        const float* bb, int lane)
{
  const int N = lane & 15;   // == A-row M for this lane
  const int hi = lane >> 4;
  float mu = 0.f, rs = 0.f;
  if (MODE == 1) { mu = y[512 + N]; rs = y[528 + N]; }
  v8f c0, c1;
  const float b0 = bias[N];
  const float b1 = bias[16 + N];
#pragma unroll
  for (int r = 0; r < 8; ++r) { c0[r] = b0; c1[r] = b1; }
#pragma unroll
  for (int kb = 0; kb < 8; ++kb) {
    const int K = kb * 4 + hi * 2;
    float a0 = y[N * 32 + K];
    float a1 = y[N * 32 + K + 1];
    if (MODE == 1) {
      a0 = __sinf((a0 - mu) * rs * g[K]     + bb[K]);
      a1 = __sinf((a1 - mu) * rs * g[K + 1] + bb[K + 1]);
    } else if (MODE == 2) {
      a0 = fmaxf(a0, 0.f);
      a1 = fmaxf(a1, 0.f);
    }
    v2f av;  av[0] = a0;                    av[1] = a1;
    v2f bv0; bv0[0] = wT[N * 32 + K];        bv0[1] = wT[N * 32 + K + 1];
    v2f bv1; bv1[0] = wT[(16 + N) * 32 + K]; bv1[1] = wT[(16 + N) * 32 + K + 1];
    c0 = __builtin_amdgcn_wmma_f32_16x16x4_f32(false, av, false, bv0, (short)0,
                                               c0, false, false);
    c1 = __builtin_amdgcn_wmma_f32_16x16x4_f32(false, av, false, bv1, (short)0,
                                               c1, false, false);
  }
  WFENCE();
#pragma unroll
  for (int r = 0; r < 8; ++r) {
    const int Mr = r + 8 * hi;
    y[Mr * 32 + N]      = c0[r];
    y[Mr * 32 + 16 + N] = c1[r];
  }
  WFENCE();
}

__device__ __forceinline__ void ln_stats(float* y, int lane)
{
  if (lane < 16) {
    const float* row = y + lane * 32;
    float s = 0.f;
#pragma unroll
    for (int k = 0; k < 32; ++k) s += row[k];
    float mu = s * (1.f / 32.f);
    float v = 0.f;
#pragma unroll
    for (int k = 0; k < 32; ++k) { float d = row[k] - mu; v += d * d; }
    v *= (1.f / 32.f);
    y[512 + lane] = mu;
    y[528 + lane] = rsqrtf(v + 1e-5f);
  }
  WFENCE();
}

// ---------------------------------------------------------------------------
// Kernel 2: full pipeline. One wave == 16 consecutive pixels of one batch.
// ---------------------------------------------------------------------------
__global__ __launch_bounds__(256) void spacetime_main(
    const float* __restrict__ t, const float* __restrict__ xyb,
    const float* __restrict__ w_in, const float* __restrict__ b_in,
    const float* __restrict__ w_h, const float* __restrict__ b_h,
    const float* __restrict__ ln_g, const float* __restrict__ ln_b,
    const float* __restrict__ w_out, const float* __restrict__ b_out,
    const float* __restrict__ r_wh, const float* __restrict__ r_bh,
    const float* __restrict__ r_wo, const float* __restrict__ r_bo,
    const float* __restrict__ feat, float* __restrict__ out)
{
  __shared__ float s_win[96];     // w_in row-major [3][32]
  __shared__ float s_bin[32];
  __shared__ float s_whT[3072];   // [layer][n][k] = w_h[layer][k][n]
  __shared__ float s_bh[96];
  __shared__ float s_lng[96];
  __shared__ float s_lnb[96];
  __shared__ float s_wox[32];     // w_out[:,0]
  __shared__ float s_woy[32];     // w_out[:,1]
  __shared__ float s_bout[2];
  __shared__ float s_rwT[3072];   // [layer][n][k] = r_wh[layer][k][n]
  __shared__ float s_rbh[96];
  __shared__ float s_rwo[32];
  __shared__ float s_rbo[1];
  __shared__ float s_y[8][608];   // per-wave: 16x32 tile + mu/rsig + samp params

  const int tid = threadIdx.x;
  // ---- stage weights into LDS (transposing the 32x32 matrices) ----
  for (int idx = tid; idx < 3072; idx += 256) {
    int l = idx >> 10, r = idx & 1023, n = r >> 5, k = r & 31;
    s_whT[idx] = w_h[l * 1024 + k * 32 + n];
    s_rwT[idx] = r_wh[l * 1024 + k * 32 + n];
  }
  for (int idx = tid; idx < 96; idx += 256) {
    s_win[idx] = w_in[idx];
    s_bh[idx] = b_h[idx];
    s_lng[idx] = ln_g[idx];
    s_lnb[idx] = ln_b[idx];
    s_rbh[idx] = r_bh[idx];
  }
  for (int idx = tid; idx < 32; idx += 256) {
    s_bin[idx] = b_in[idx];
    s_wox[idx] = w_out[2 * idx + 0];
    s_woy[idx] = w_out[2 * idx + 1];
    s_rwo[idx] = r_wo[idx];
  }
  if (tid == 0) { s_bout[0] = b_out[0]; s_bout[1] = b_out[1]; s_rbo[0] = r_bo[0]; }
  __syncthreads();

  const int lane = tid & 31;
  const int wave = tid >> 5;
  const int s = blockIdx.x * 8 + wave;        // global 16-pixel strip id
  const int b = s >> 14;                      // batch (16384 strips / batch)
  const int strip = s & 16383;
  const int p_base = strip << 4;              // first pixel of strip
  const int i = p_base >> 9;                  // row (X index)
  const int j_base = p_base & 511;            // col base (Y index)

  const int M = lane & 15;
  const int hi = lane >> 4;
  float* y = s_y[wave];

  const int j = j_base + M;
  const float gx = xyb[((size_t)(i * XDIM + j)) * 2 + 0];
  const float gy = xyb[((size_t)(i * XDIM + j)) * 2 + 1];
  const float alpha = t[b] + 0.5f;

  // ---- input layer h0 = [gx,gy,alpha] @ w_in + b_in ----
#pragma unroll
  for (int c2 = 0; c2 < 16; ++c2) {
    int f = hi * 16 + c2;
    y[M * 32 + f] = gx * s_win[f] + gy * s_win[32 + f] + alpha * s_win[64 + f]
                    + s_bin[f];
  }
  WFENCE();

  // ---- warp MLP: 3x (matmul -> LN -> sin), activation fused into A-load ----
  layer32<0>(y, s_whT + 0,    s_bh + 0,  nullptr,    nullptr,    lane);
  ln_stats(y, lane);
  layer32<1>(y, s_whT + 1024, s_bh + 32, s_lng + 0,  s_lnb + 0,  lane);
  ln_stats(y, lane);
  layer32<1>(y, s_whT + 2048, s_bh + 64, s_lng + 32, s_lnb + 32, lane);
  ln_stats(y, lane);

  // ---- motion head (32->2) + sampling coordinates ----
  if (lane < 16) {
    float mu = y[512 + lane], rs = y[528 + lane];
    float mx = s_bout[0], my = s_bout[1];
#pragma unroll
    for (int k = 0; k < 32; ++k) {
      float a = __sinf((y[lane * 32 + k] - mu) * rs * s_lng[64 + k]
                       + s_lnb[64 + k]);
      mx += a * s_wox[k];
      my += a * s_woy[k];
    }
    float gxg = gx + 0.1f * mx;
    float gyg = gy + 0.1f * my;
    float ix = ((gxg + 1.f) * (float)XDIM - 1.f) * 0.5f;
    float iy = ((gyg + 1.f) * (float)XDIM - 1.f) * 0.5f;
    float xf = floorf(ix), yf = floorf(iy);
    y[544 + lane * 4 + 0] = xf;
    y[544 + lane * 4 + 1] = yf;
    y[544 + lane * 4 + 2] = ix - xf;
    y[544 + lane * 4 + 3] = iy - yf;
  }
  WFENCE();

  // ---- grid_sample (zero padding): lane handles pixel M, channels hi*16.. ----
  {
    float xf = y[544 + M * 4 + 0];
    float yf = y[544 + M * 4 + 1];
    float wx = y[544 + M * 4 + 2];
    float wy = y[544 + M * 4 + 3];
    float acc[16];
#pragma unroll
    for (int c2 = 0; c2 < 16; ++c2) acc[c2] = 0.f;
#pragma unroll
    for (int cy = 0; cy < 2; ++cy) {
      float yc = yf + (float)cy;
      float wyc = cy ? wy : (1.f - wy);
#pragma unroll
      for (int cx = 0; cx < 2; ++cx) {
        float xc = xf + (float)cx;
        float wxc = cx ? wx : (1.f - wx);
        float w = wxc * wyc;
        if (xc >= 0.f && xc <= (float)(XDIM - 1) && yc >= 0.f &&
            yc <= (float)(XDIM - 1)) {
          int xi = (int)xc, yi = (int)yc;
          const float4* src =
              (const float4*)(feat + ((size_t)yi * XDIM + xi) * CDIM + hi * 16);
#pragma unroll
          for (int q = 0; q < 4; ++q) {
            float4 v = src[q];
            acc[q * 4 + 0] += w * v.x;
            acc[q * 4 + 1] += w * v.y;
            acc[q * 4 + 2] += w * v.z;
            acc[q * 4 + 3] += w * v.w;
          }
        }
      }
    }
    WFENCE();
#pragma unroll
    for (int c2 = 0; c2 < 16; ++c2) y[M * 32 + hi * 16 + c2] = acc[c2];
  }
  WFENCE();

  // ---- render MLP: 3x (matmul -> relu), relu fused into next A-load ----
  layer32<0>(y, s_rwT + 0,    s_rbh + 0,  nullptr, nullptr, lane);
  layer32<2>(y, s_rwT + 1024, s_rbh + 32, nullptr, nullptr, lane);
  layer32<2>(y, s_rwT + 2048, s_rbh + 64, nullptr, nullptr, lane);

  // ---- output head (32->1) + leaky relu ----
  if (lane < 16) {
    float o = s_rbo[0];
#pragma unroll
    for (int k = 0; k < 32; ++k)
      o += fmaxf(y[lane * 32 + k], 0.f) * s_rwo[k];
    o = (o >= 0.f) ? o : 0.001f * o;
    out[(size_t)b * NPIX + p_base + lane] = o;
  }
}

extern "C" void kernel_launch(void* const* d_in, const int* in_sizes, int n_in,
                              void* d_out, int out_size, void* d_ws,
                              size_t ws_size, hipStream_t stream)
{
  const float* t      = (const float*)d_in[0];
  const float* data   = (const float*)d_in[1];
  const float* lerp_w = (const float*)d_in[2];
  const float* xyb    = (const float*)d_in[3];
  const float* w_in   = (const float*)d_in[4];
  const float* b_in   = (const float*)d_in[5];
  const float* w_h    = (const float*)d_in[6];
  const float* b_h    = (const float*)d_in[7];
  const float* ln_g   = (const float*)d_in[8];
  const float* ln_b   = (const float*)d_in[9];
  const float* w_out  = (const float*)d_in[10];
  const float* b_out  = (const float*)d_in[11];
  const float* r_wh   = (const float*)d_in[12];
  const float* r_bh   = (const float*)d_in[13];
  const float* r_wo   = (const float*)d_in[14];
  const float* r_bo   = (const float*)d_in[15];
  const int* x0       = (const int*)d_in[16];
  const int* y0       = (const int*)d_in[17];
  const int* x1       = (const int*)d_in[18];
  const int* y1       = (const int*)d_in[19];

  float* feat = (float*)d_ws;          // needs 512*512*32*4 = 32 MB
  float* out  = (float*)d_out;

  feat_precompute<<<NPIX / 256, 256, 0, stream>>>(data, lerp_w, x0, y0, x1, y1,
                                                  feat);

  // 8 batches * 16384 strips of 16 pixels; 8 waves (strips) per block
  spacetime_main<<<16384, 256, 0, stream>>>(t, xyb, w_in, b_in, w_h, b_h, ln_g,
                                            ln_b, w_out, b_out, r_wh, r_bh,
                                            r_wo, r_bo, feat, out);
}